// ComplexMLP_15496242004669
// MI455X (gfx1250) — compile-verified
//
#include <hip/hip_runtime.h>
#include <cmath>

typedef __attribute__((ext_vector_type(2))) float v2f;
typedef __attribute__((ext_vector_type(8))) float v8f;

#define DT_C 0.001f
#define STEPS_C 64
#define U_C 1024
#define A_C 512
#define B_C 128

// ---- GEMM tile config ----
#define BM 64
#define BN 128
#define BK 32
#define LDA_S (BK + 4)   // 36: float4-aligned padded stride for A tiles
#define LDB_S (BN + 4)   // 132: float4-aligned padded stride for B tiles

// Complex GEMM with tanh epilogue.
//  Yr = tanh(Xr@Wr - Xi@Wi + br), Yi = tanh(Xr@Wi + Xi@Wr + bi)
// HAS_XI : Xi present (else Xi treated as zero)
// WANT_YI: compute/store imaginary output
// ACC    : Y += tanh(...)  (for summing the two tanh'd cdenses of rnn())
// REMAP  : store Yr with row r -> out[(r&127)*STEPS*A + (r>>7)*A + col]
template <bool HAS_XI, bool WANT_YI, bool ACC, bool REMAP>
__global__ __launch_bounds__(256) void cgemm_tanh(
    const float* __restrict__ Xr, const float* __restrict__ Xi, int ldx,
    const float* __restrict__ Wr, const float* __restrict__ Wi,
    const float* __restrict__ br, const float* __restrict__ bi,
    float* __restrict__ Yr, float* __restrict__ Yi, int ldy,
    int N, int K, int rowOfs)
{
    __shared__ float lAr[BM * LDA_S];
    __shared__ float lAi[BM * LDA_S];
    __shared__ float lBr[BK * LDB_S];
    __shared__ float lBi[BK * LDB_S];

    const int tid  = threadIdx.x;
    const int lane = tid & 31;
    const int wave = tid >> 5;
    const int wm   = wave & 1;      // 2 M-groups of 32 rows
    const int wn   = wave >> 1;     // 4 N-groups of 32 cols
    const int hi   = lane >> 4;     // half-wave select (K phase / M phase)
    const int l15  = lane & 15;

    const int blockM = blockIdx.y * BM;
    const int blockN = blockIdx.x * BN;

    v8f acc_r[2][2] = {};
    v8f acc_i[2][2] = {};

    for (int k0 = 0; k0 < K; k0 += BK) {
        // ---- stage A tiles (BM x BK) ----
        #pragma unroll
        for (int i = 0; i < 2; ++i) {
            int j  = tid + i * 256;           // 512 float4 slots
            int r  = j >> 3;
            int c4 = (j & 7) << 2;
            const float4 ar4 = *(const float4*)(Xr + (size_t)(blockM + r) * ldx + k0 + c4);
            *(float4*)&lAr[r * LDA_S + c4] = ar4;
            if constexpr (HAS_XI) {
                const float4 ai4 = *(const float4*)(Xi + (size_t)(blockM + r) * ldx + k0 + c4);
                *(float4*)&lAi[r * LDA_S + c4] = ai4;
            }
        }
        // ---- stage B tiles (BK x BN) ----
        #pragma unroll
        for (int i = 0; i < 4; ++i) {
            int j  = tid + i * 256;           // 1024 float4 slots
            int r  = j >> 5;
            int c4 = (j & 31) << 2;
            const float4 br4 = *(const float4*)(Wr + (size_t)(k0 + r) * N + blockN + c4);
            *(float4*)&lBr[r * LDB_S + c4] = br4;
            if constexpr (WANT_YI || HAS_XI) {
                const float4 bi4 = *(const float4*)(Wi + (size_t)(k0 + r) * N + blockN + c4);
                *(float4*)&lBi[r * LDB_S + c4] = bi4;
            }
        }
        __syncthreads();

        #pragma unroll
        for (int kk = 0; kk < BK; kk += 4) {
            v2f ar[2], ai[2], bre[2], bim[2];
            #pragma unroll
            for (int mi = 0; mi < 2; ++mi) {
                // 32-bit A layout: lanes 0-15 hold (M=l, K=kk..kk+1), lanes 16-31 (M=l, K=kk+2..kk+3)
                int rowa = (wm * 32 + mi * 16 + l15) * LDA_S + kk + hi * 2;
                ar[mi].x = lAr[rowa];
                ar[mi].y = lAr[rowa + 1];
                if constexpr (HAS_XI) { ai[mi].x = lAi[rowa]; ai[mi].y = lAi[rowa + 1]; }
            }
            #pragma unroll
            for (int ni = 0; ni < 2; ++ni) {
                // B layout mirrors A: VGPR0 = row K=kk (+2 for hi), VGPR1 = next K row
                int colb = wn * 32 + ni * 16 + l15;
                int rb0  = (kk + hi * 2) * LDB_S + colb;
                bre[ni].x = lBr[rb0];
                bre[ni].y = lBr[rb0 + LDB_S];
                if constexpr (WANT_YI || HAS_XI) {
                    bim[ni].x = lBi[rb0];
                    bim[ni].y = lBi[rb0 + LDB_S];
                }
            }
            #pragma unroll
            for (int ni = 0; ni < 2; ++ni) {
                v2f bin;
                if constexpr (HAS_XI) bin = -bim[ni];  // f32 WMMA has no A/B NEG bits; negate in VALU
                #pragma unroll
                for (int mi = 0; mi < 2; ++mi) {
                    acc_r[mi][ni] = __builtin_amdgcn_wmma_f32_16x16x4_f32(
                        false, ar[mi], false, bre[ni], (short)0, acc_r[mi][ni], false, false);
                    if constexpr (HAS_XI)
                        acc_r[mi][ni] = __builtin_amdgcn_wmma_f32_16x16x4_f32(
                            false, ai[mi], false, bin, (short)0, acc_r[mi][ni], false, false);
                    if constexpr (WANT_YI) {
                        acc_i[mi][ni] = __builtin_amdgcn_wmma_f32_16x16x4_f32(
                            false, ar[mi], false, bim[ni], (short)0, acc_i[mi][ni], false, false);
                        if constexpr (HAS_XI)
                            acc_i[mi][ni] = __builtin_amdgcn_wmma_f32_16x16x4_f32(
                                false, ai[mi], false, bre[ni], (short)0, acc_i[mi][ni], false, false);
                    }
                }
            }
        }
        __syncthreads();
    }

    // ---- epilogue: bias + tanh (+accumulate) ----
    #pragma unroll
    for (int mi = 0; mi < 2; ++mi) {
        #pragma unroll
        for (int ni = 0; ni < 2; ++ni) {
            int coln  = blockN + wn * 32 + ni * 16 + l15;
            float biasr = br[coln];
            float biasi = 0.0f;
            if constexpr (WANT_YI) biasi = bi[coln];
            int rbase = blockM + wm * 32 + mi * 16 + hi * 8;
            #pragma unroll
            for (int j = 0; j < 8; ++j) {
                int r = rbase + j;
                float vr = tanhf(acc_r[mi][ni][j] + biasr);
                if constexpr (REMAP) {
                    int gr = rowOfs + r;
                    int t  = gr >> 7;         // step index
                    int bb = gr & 127;        // batch index
                    size_t oidx = (size_t)bb * (STEPS_C * A_C) + (size_t)t * A_C + coln;
                    Yr[oidx] = vr;
                } else {
                    size_t oidx = (size_t)r * ldy + coln;
                    if constexpr (ACC) Yr[oidx] += vr; else Yr[oidx] = vr;
                    if constexpr (WANT_YI) {
                        float vi = tanhf(acc_i[mi][ni][j] + biasi);
                        if constexpr (ACC) Yi[oidx] += vi; else Yi[oidx] = vi;
                    }
                }
            }
        }
    }
}

// real = concat(robot_state, motion_state) -> [128,1024]
__global__ void concat_real_kernel(const float* __restrict__ robot,
                                   const float* __restrict__ motion,
                                   float* __restrict__ xrbuf)
{
    int idx = blockIdx.x * blockDim.x + threadIdx.x;  // 131072
    int b = idx >> 10, u = idx & 1023;
    float v = (u < 512) ? robot[b * 512 + u] : motion[b * 512 + (u - 512)];
    xrbuf[idx] = v;
}

// Run all 64 Hopf Euler steps; z1 -> z1buf and d_out tail; z_t (t>=2) -> XB row blocks 1..63.
__global__ void hopf_all_kernel(const float* __restrict__ z,
                                const float* __restrict__ omega,
                                float* __restrict__ z1buf,
                                float* __restrict__ z1out,
                                float* __restrict__ XB)
{
    int idx = blockIdx.x * blockDim.x + threadIdx.x;  // 131072
    int b = idx >> 10, u = idx & 1023;
    float x = z[b * 2048 + u];
    float y = z[b * 2048 + 1024 + u];
    float w = omega[b] * (float)(u + 1);
    #pragma unroll 1
    for (int t = 1; t <= STEPS_C; ++t) {
        float r2 = x * x + y * y;
        float xn = x + DT_C * ((1.0f - r2) * x - w * y);
        float yn = y + DT_C * ((1.0f - r2) * y + w * x);
        x = xn; y = yn;
        if (t == 1) {
            z1buf[b * 2048 + u] = x;
            z1buf[b * 2048 + 1024 + u] = y;
            z1out[b * 2048 + u] = x;
            z1out[b * 2048 + 1024 + u] = y;
        } else {
            size_t r = (size_t)(t - 1) * B_C + b;   // row block t-1 holds z_t
            XB[r * 2048 + u] = x;
            XB[r * 2048 + 1024 + u] = y;
        }
    }
}

extern "C" void kernel_launch(void* const* d_in, const int* in_sizes, int n_in,
                              void* d_out, int out_size, void* d_ws, size_t ws_size,
                              hipStream_t stream) {
    const float* z       = (const float*)d_in[0];
    const float* omega   = (const float*)d_in[1];
    const float* robot   = (const float*)d_in[2];
    const float* motion  = (const float*)d_in[3];
    const float* cw_re   = (const float*)d_in[4];
    const float* cw_im   = (const float*)d_in[5];
    const float* cb_re   = (const float*)d_in[6];
    const float* cb_im   = (const float*)d_in[7];
    const float* ow_re   = (const float*)d_in[8];
    const float* ow_im   = (const float*)d_in[9];
    const float* ob_re   = (const float*)d_in[10];
    const float* ob_im   = (const float*)d_in[11];
    const float* sw_re   = (const float*)d_in[12];
    const float* sw_im   = (const float*)d_in[13];
    const float* sb_re   = (const float*)d_in[14];
    const float* sb_im   = (const float*)d_in[15];
    const float* m1w_re  = (const float*)d_in[16];
    const float* m1w_im  = (const float*)d_in[17];
    const float* m1b_re  = (const float*)d_in[18];
    const float* m1b_im  = (const float*)d_in[19];
    const float* m2w_re  = (const float*)d_in[20];
    const float* m2w_im  = (const float*)d_in[21];
    const float* m2b_re  = (const float*)d_in[22];
    // m2b_im (d_in[23]) unused: only the real half of the final dense survives [:, :, :A]

    float* out    = (float*)d_out;
    float* z1out  = out + (size_t)B_C * STEPS_C * A_C;   // second tuple output

    // workspace layout (floats)
    float* ws     = (float*)d_ws;
    float* xrbuf  = ws;                                  // 128*1024
    float* s0     = xrbuf + 131072;                      // 128*2048 (r|i)
    float* z1buf  = s0 + 262144;                         // 128*2048 (r|i)
    float* XB     = z1buf + 262144;                      // 8192*2048 (r|i): row block 0 = o_rnn0, blocks 1..63 = z_2..z_64
    float* H1     = XB + (size_t)8192 * 2048;            // 1024*4096 chunk (r|i)

    // 1) real input concat + 2) all Hopf steps
    concat_real_kernel<<<512, 256, 0, stream>>>(robot, motion, xrbuf);
    hopf_all_kernel<<<512, 256, 0, stream>>>(z, omega, z1buf, z1out, XB);

    dim3 blk(256);
    dim3 gSmall(1024 / BN, B_C / BM);   // M=128, N=1024

    // 3) state0 = [tanh(xr@cw_re+cb_re) | tanh(xr@cw_im+cb_im)]  (xi = 0)
    cgemm_tanh<false, true, false, false><<<gSmall, blk, 0, stream>>>(
        xrbuf, nullptr, 1024, cw_re, cw_im, cb_re, cb_im,
        s0, s0 + 1024, 2048, 1024, 1024, 0);

    // 4) o_rnn0 pass 1: tanh(cdense(z1, ow)) -> XB rows 0..127
    cgemm_tanh<true, true, false, false><<<gSmall, blk, 0, stream>>>(
        z1buf, z1buf + 1024, 2048, ow_re, ow_im, ob_re, ob_im,
        XB, XB + 1024, 2048, 1024, 1024, 0);

    // 5) o_rnn0 pass 2: += tanh(cdense(state0, sw))
    cgemm_tanh<true, true, true, false><<<gSmall, blk, 0, stream>>>(
        s0, s0 + 1024, 2048, sw_re, sw_im, sb_re, sb_im,
        XB, XB + 1024, 2048, 1024, 1024, 0);

    // 6) batched MLP over all 64 step-blocks (8192 rows), chunks of 1024 rows
    for (int c = 0; c < 8; ++c) {
        const float* Xc = XB + (size_t)c * 1024 * 2048;
        dim3 g3(2048 / BN, 1024 / BM);
        cgemm_tanh<true, true, false, false><<<g3, blk, 0, stream>>>(
            Xc, Xc + 1024, 2048, m1w_re, m1w_im, m1b_re, m1b_im,
            H1, H1 + 2048, 4096, 2048, 1024, 0);

        dim3 g4(512 / BN, 1024 / BM);
        cgemm_tanh<true, false, false, true><<<g4, blk, 0, stream>>>(
            H1, H1 + 2048, 4096, m2w_re, m2w_im, m2b_re, nullptr,
            out, nullptr, 0, 512, 2048, c * 1024);
    }
}